// GumbelSampler_62440234549893
// MI455X (gfx1250) — compile-verified
//
#include <hip/hip_runtime.h>
#include <stdint.h>

#define NMAX       8192
#define NROWS      2048          // repeat(2) * bsz(256) * ensemble(4)
#define THREADS    256
#define PER_THREAD (NMAX / THREADS)   // 32 elements per thread
#define KSEL       32
#define TAU_INV    10.0f
#define EPS_TINY   1.1754943508222875e-38f  // np.finfo(float32).tiny

typedef unsigned int v4u __attribute__((ext_vector_type(4)));
typedef int          v4i __attribute__((ext_vector_type(4)));
typedef int          v8i __attribute__((ext_vector_type(8)));

#if defined(__has_builtin)
#  if __has_builtin(__builtin_amdgcn_tensor_load_to_lds) && \
      __has_builtin(__builtin_amdgcn_tensor_store_from_lds) && \
      __has_builtin(__builtin_amdgcn_s_wait_tensorcnt)
#    define HAVE_TDM 1
#  endif
#endif
#ifndef HAVE_TDM
#  define HAVE_TDM 0
#endif

// ---------------- Threefry2x32 (JAX-compatible) ----------------
__device__ __forceinline__ uint32_t rotl32(uint32_t v, int r) {
  return (v << r) | (v >> (32 - r));
}

__device__ __forceinline__ void threefry2x32(uint32_t k0, uint32_t k1,
                                             uint32_t& x0, uint32_t& x1) {
  const uint32_t ks2 = k0 ^ k1 ^ 0x1BD11BDAu;
  x0 += k0; x1 += k1;
  x0 += x1; x1 = rotl32(x1, 13); x1 ^= x0;
  x0 += x1; x1 = rotl32(x1, 15); x1 ^= x0;
  x0 += x1; x1 = rotl32(x1, 26); x1 ^= x0;
  x0 += x1; x1 = rotl32(x1,  6); x1 ^= x0;
  x0 += k1; x1 += ks2 + 1u;
  x0 += x1; x1 = rotl32(x1, 17); x1 ^= x0;
  x0 += x1; x1 = rotl32(x1, 29); x1 ^= x0;
  x0 += x1; x1 = rotl32(x1, 16); x1 ^= x0;
  x0 += x1; x1 = rotl32(x1, 24); x1 ^= x0;
  x0 += ks2; x1 += k0 + 2u;
  x0 += x1; x1 = rotl32(x1, 13); x1 ^= x0;
  x0 += x1; x1 = rotl32(x1, 15); x1 ^= x0;
  x0 += x1; x1 = rotl32(x1, 26); x1 ^= x0;
  x0 += x1; x1 = rotl32(x1,  6); x1 ^= x0;
  x0 += k0; x1 += k1 + 3u;
  x0 += x1; x1 = rotl32(x1, 17); x1 ^= x0;
  x0 += x1; x1 = rotl32(x1, 29); x1 ^= x0;
  x0 += x1; x1 = rotl32(x1, 16); x1 ^= x0;
  x0 += x1; x1 = rotl32(x1, 24); x1 ^= x0;
  x0 += k1; x1 += ks2 + 4u;
  x0 += x1; x1 = rotl32(x1, 13); x1 ^= x0;
  x0 += x1; x1 = rotl32(x1, 15); x1 ^= x0;
  x0 += x1; x1 = rotl32(x1, 26); x1 ^= x0;
  x0 += x1; x1 = rotl32(x1,  6); x1 ^= x0;
  x0 += ks2; x1 += k0 + 5u;
}

// JAX gumbel(key=42) at flat index j of a (NROWS, NMAX) array.
__device__ __forceinline__ float gumbel_at(uint32_t j) {
  const uint32_t HALF = (uint32_t)NROWS * (uint32_t)NMAX / 2u;  // 8388608
  uint32_t c0, c1;
  const bool lo = (j < HALF);
  if (lo) { c0 = j;        c1 = j + HALF; }
  else    { c0 = j - HALF; c1 = j;        }
  threefry2x32(0u, 42u, c0, c1);
  const uint32_t bits = lo ? c0 : c1;
  float f = __uint_as_float((bits >> 9) | 0x3F800000u) - 1.0f;
  float u = f * (1.0f - EPS_TINY) + EPS_TINY;
  u = fmaxf(u, EPS_TINY);
  return -__logf(-__logf(u));
}

// ---------------- TDM descriptor: 1 x 8192 tile, 4B elems, stride 4 elems ----
#if HAVE_TDM
__device__ __forceinline__ void make_desc(uint64_t gaddr_bytes, uint32_t lds_off,
                                          v4u& g0, v8i& g1) {
  g0.x = 1u;                                        // count=1, user descriptor
  g0.y = lds_off;                                   // LDS byte address
  g0.z = (uint32_t)gaddr_bytes;                     // global addr [31:0]
  g0.w = (uint32_t)(gaddr_bytes >> 32) | (2u << 30);// addr[56:32] | type=2
  g1[0] = (int)(2u << 16);       // workgroup_mask=0, data_size=2 (4 bytes)
  g1[1] = (int)(1u << 16);       // atomic_barrier=0, tensor_dim0 lo16 = 1
  g1[2] = (int)(8192u << 16);    // tensor_dim0 hi16=0, tensor_dim1 lo16 = 8192
  g1[3] = (int)(1u << 16);       // tensor_dim1 hi16=0, tile_dim0 = 1
  g1[4] = (int)8192;             // tile_dim1 = 8192, tile_dim2 = 0
  g1[5] = 4;                     // tensor_dim0_stride = 4 elements (ensemble)
  g1[6] = 0;                     // stride0 hi, stride1 lo
  g1[7] = 0;                     // stride1 hi
}

__device__ __forceinline__ void tdm_issue_load(v4u g0, v8i g1) {
  v4i z4 = {0, 0, 0, 0};
#if __clang_major__ >= 23
  v8i z8 = {0, 0, 0, 0, 0, 0, 0, 0};
  __builtin_amdgcn_tensor_load_to_lds(g0, g1, z4, z4, z8, 0);
#else
  __builtin_amdgcn_tensor_load_to_lds(g0, g1, z4, z4, 0);
#endif
}

__device__ __forceinline__ void tdm_issue_store(v4u g0, v8i g1) {
  v4i z4 = {0, 0, 0, 0};
#if __clang_major__ >= 23
  v8i z8 = {0, 0, 0, 0, 0, 0, 0, 0};
  __builtin_amdgcn_tensor_store_from_lds(g0, g1, z4, z4, z8, 0);
#else
  __builtin_amdgcn_tensor_store_from_lds(g0, g1, z4, z4, 0);
#endif
}
#endif  // HAVE_TDM

// ---------------- Block reductions (8 wave32 waves) ----------------
__device__ __forceinline__ float block_max_f(float v, float* sm, int t) {
  #pragma unroll
  for (int m = 16; m > 0; m >>= 1) v = fmaxf(v, __shfl_xor(v, m, 32));
  __syncthreads();
  if ((t & 31) == 0) sm[t >> 5] = v;
  __syncthreads();
  float r = sm[0];
  #pragma unroll
  for (int i = 1; i < 8; ++i) r = fmaxf(r, sm[i]);
  return r;
}

__device__ __forceinline__ float block_sum_f(float v, float* sm, int t) {
  #pragma unroll
  for (int m = 16; m > 0; m >>= 1) v += __shfl_xor(v, m, 32);
  __syncthreads();
  if ((t & 31) == 0) sm[t >> 5] = v;
  __syncthreads();
  float r = sm[0];
  #pragma unroll
  for (int i = 1; i < 8; ++i) r += sm[i];
  return r;
}

__device__ __forceinline__ unsigned long long
block_max_u64(unsigned long long v, unsigned long long* sm, int t) {
  #pragma unroll
  for (int m = 16; m > 0; m >>= 1) {
    unsigned long long o = __shfl_xor(v, m, 32);
    v = (o > v) ? o : v;
  }
  __syncthreads();
  if ((t & 31) == 0) sm[t >> 5] = v;
  __syncthreads();
  unsigned long long r = sm[0];
  #pragma unroll
  for (int i = 1; i < 8; ++i) r = (sm[i] > r) ? sm[i] : r;
  return r;
}

// ---------------- Main kernel: one row per block ----------------
__global__ __launch_bounds__(THREADS)
void gumbel_topk_kernel(const float* __restrict__ scores, float* __restrict__ out) {
  const int r   = blockIdx.x;            // 0..2047
  const int rep = r >> 10;
  const int be  = r & 1023;
  const int b   = be >> 2;
  const int e   = be & 3;
  const int t   = threadIdx.x;

  __shared__ float              row[NMAX];   // 32KB staging (TDM src/dst)
  __shared__ float              sm_f[8];
  __shared__ unsigned long long sm_u[8];

  float f[PER_THREAD];
  float khot[PER_THREAD];

  // ---- stage score row (stride 16B gather) via Tensor Data Mover ----
#if HAVE_TDM
  const uint64_t in_addr = (uint64_t)(uintptr_t)scores +
                           (((uint64_t)b * NMAX) * 4u + (uint64_t)e) * 4u;
  if (t == 0) {
    v4u g0; v8i g1;
    make_desc(in_addr, (uint32_t)(uintptr_t)(void*)row, g0, g1);
    tdm_issue_load(g0, g1);
    __builtin_amdgcn_s_wait_tensorcnt(0);
  }
  __syncthreads();
#else
  const float* __restrict__ srow = scores + ((size_t)b * NMAX) * 4 + e;
#endif

  #pragma unroll
  for (int i = 0; i < PER_THREAD; ++i) {
    const int n = i * THREADS + t;
#if HAVE_TDM
    const float sc = row[n];               // conflict-free ds_load
#else
    if (i + 8 < PER_THREAD)
      __builtin_prefetch(&srow[(size_t)((i + 8) * THREADS + t) * 4], 0, 3);
    const float sc = srow[(size_t)n * 4];
#endif
    f[i]    = sc + gumbel_at((uint32_t)r * (uint32_t)NMAX + (uint32_t)n);
    khot[i] = 0.0f;
  }

  // ---- 32 relaxed-topk iterations; onehot recomputed from (m, 1/s) ----
  float m_prev = 0.0f, inv_s_prev = 0.0f;
  for (int it = 0; it < KSEL; ++it) {
    if (it > 0) {
      #pragma unroll
      for (int i = 0; i < PER_THREAD; ++i) {
        const float one = __expf((f[i] - m_prev) * TAU_INV) * inv_s_prev;
        khot[i] += one;
        f[i]    += __logf(fmaxf(1.0f - one, EPS_TINY));
      }
    }
    float lm = -3.4e38f;
    #pragma unroll
    for (int i = 0; i < PER_THREAD; ++i) lm = fmaxf(lm, f[i]);
    lm = block_max_f(lm, sm_f, t);

    float ls = 0.0f;
    #pragma unroll
    for (int i = 0; i < PER_THREAD; ++i) ls += __expf((f[i] - lm) * TAU_INV);
    ls = block_sum_f(ls, sm_f, t);

    m_prev = lm; inv_s_prev = 1.0f / ls;
  }
  #pragma unroll
  for (int i = 0; i < PER_THREAD; ++i)
    khot[i] += __expf((f[i] - m_prev) * TAU_INV) * inv_s_prev;

  // ---- hard top-32: iterative packed argmax (ties -> lower index) ----
  uint32_t hard = 0u;
  for (int sel = 0; sel < KSEL; ++sel) {
    unsigned long long best = 0ull;
    #pragma unroll
    for (int i = 0; i < PER_THREAD; ++i) {
      if (!((hard >> i) & 1u)) {
        const uint32_t n = (uint32_t)(i * THREADS + t);
        const unsigned long long pk =
            ((unsigned long long)__float_as_uint(khot[i]) << 32) |
            (unsigned long long)(~n);
        best = (pk > best) ? pk : best;
      }
    }
    best = block_max_u64(best, sm_u, t);
    const uint32_t n_sel = ~(uint32_t)(best & 0xFFFFFFFFull);
    if ((int)(n_sel & (THREADS - 1)) == t) hard |= 1u << (n_sel >> 8);
  }

  // ---- straight-through output, layout (rep, b, n, e) ----
#if HAVE_TDM
  __syncthreads();
  #pragma unroll
  for (int i = 0; i < PER_THREAD; ++i) {
    const int n = i * THREADS + t;
    const float h = ((hard >> i) & 1u) ? 1.0f : 0.0f;
    row[n] = (h - khot[i]) + khot[i];
  }
  __syncthreads();
  if (t == 0) {
    const uint64_t out_addr = (uint64_t)(uintptr_t)out +
        (((uint64_t)(rep * 256 + b) * NMAX) * 4u + (uint64_t)e) * 4u;
    v4u g0; v8i g1;
    make_desc(out_addr, (uint32_t)(uintptr_t)(void*)row, g0, g1);
    tdm_issue_store(g0, g1);
    __builtin_amdgcn_s_wait_tensorcnt(0);
  }
#else
  float* __restrict__ orow = out + ((size_t)(rep * 256 + b) * NMAX) * 4 + e;
  #pragma unroll
  for (int i = 0; i < PER_THREAD; ++i) {
    const int n = i * THREADS + t;
    const float h = ((hard >> i) & 1u) ? 1.0f : 0.0f;
    orow[(size_t)n * 4] = (h - khot[i]) + khot[i];
  }
#endif
}

extern "C" void kernel_launch(void* const* d_in, const int* in_sizes, int n_in,
                              void* d_out, int out_size, void* d_ws, size_t ws_size,
                              hipStream_t stream) {
  const float* scores = (const float*)d_in[0];   // (256, 8192, 4) f32
  float* out = (float*)d_out;                    // (2, 256, 8192, 4) f32
  (void)in_sizes; (void)n_in; (void)out_size; (void)d_ws; (void)ws_size;
  gumbel_topk_kernel<<<NROWS, THREADS, 0, stream>>>(scores, out);
}